// KUNi_34067680591909
// MI455X (gfx1250) — compile-verified
//
#include <hip/hip_runtime.h>

typedef __attribute__((ext_vector_type(2))) float v2f;
typedef __attribute__((ext_vector_type(8))) float v8f;

#define NB    4
#define NPTS  8192
#define NTILE (NPTS / 16)          // 512 tiles per dim
#define NPTOT (NB * NPTS)          // 32768 points
#define LAMBDA 0.001f

// ---------------------------------------------------------------------------
// Pass 1: per point, pack {x, y, z, |c|^2} (16B aligned) and w = sigmoid(in1-in0).
// Also zeroes the output scalar.
// ---------------------------------------------------------------------------
__global__ __launch_bounds__(256) void KUN_dpp_pack(
    const float* __restrict__ inputs,   // (B, 2, N)
    const float* __restrict__ coords,   // (B, N, 3)
    float4* __restrict__ packed,        // (B*N) {x,y,z,sq}
    float*  __restrict__ wbuf,          // (B*N) weights
    float*  __restrict__ out)
{
    const int gid = blockIdx.x * blockDim.x + threadIdx.x;
    if (gid == 0) out[0] = 0.0f;
    if (gid >= NPTOT) return;

    const int b = gid / NPTS;
    const int n = gid - b * NPTS;

    const float x = coords[(size_t)gid * 3 + 0];
    const float y = coords[(size_t)gid * 3 + 1];
    const float z = coords[(size_t)gid * 3 + 2];

    const float i0 = inputs[(size_t)b * 2 * NPTS + n];           // channel 0
    const float i1 = inputs[(size_t)b * 2 * NPTS + NPTS + n];    // channel 1

    float4 p;
    p.x = x; p.y = y; p.z = z;
    p.w = x * x + y * y + z * z;
    packed[gid] = p;
    // sigmoid(i1 - i0) = 1 / (1 + exp(i0 - i1)) : v_exp_f32 + v_rcp_f32
    wbuf[gid] = __builtin_amdgcn_rcpf(1.0f + __expf(i0 - i1));
}

// One 16x16 Gram tile via V_WMMA_F32_16X16X4_F32 (K=3 zero-padded to 4),
// returns wm * sum_v( sqrt(max(sqn+sqm-2G,0)) * wn[v] ) for this lane.
__device__ __forceinline__ float KUN_tile(
    const float4* __restrict__ pb, const float* __restrict__ wb,
    v2f av, const float* sqn, const float* wn, int mt, int l16, int hi)
{
    const int col = mt * 16 + l16;
    const float4 cp = pb[col];                                 // b128, aligned
    const float  wm = wb[col];                                 // b32
    v2f bv;
    bv.x = hi ? cp.z : cp.x;
    bv.y = hi ? 0.0f : cp.y;

    v8f g = {};
    g = __builtin_amdgcn_wmma_f32_16x16x4_f32(
            /*neg_a=*/false, av, /*neg_b=*/false, bv,
            /*c_mod=*/(short)0, g, /*reuse_a=*/false, /*reuse_b=*/false);

    float colsum = 0.0f;
#pragma unroll
    for (int v = 0; v < 8; ++v) {
        const float d2 = (sqn[v] + cp.w) - 2.0f * g[v];
        const float d  = __builtin_amdgcn_sqrtf(fmaxf(d2, 0.0f));  // v_sqrt_f32
        colsum = fmaf(d, wn[v], colsum);
    }
    return wm * colsum;
}

// ---------------------------------------------------------------------------
// Pass 2 (triangular): each wave owns row-tile pair (p, 511-p) and one column
// parity; per row tile it sweeps only m-tiles >= row tile. Diagonal tiles get
// weight 1 (acc1), strictly-upper tiles weight 2 (acc2). Perfectly balanced:
// every wave does ~256-257 tiles, half the work of the full sweep.
// ---------------------------------------------------------------------------
__global__ __launch_bounds__(256) void KUN_dpp_kernel(
    const float4* __restrict__ packed,  // (B*N) {x,y,z,sq}
    const float*  __restrict__ wbuf,    // (B*N)
    float* __restrict__ out)
{
    const int lane = threadIdx.x & 31;
    // Force wave-uniform control values into SGPRs so all branches/loop bounds
    // are scalar (EXEC stays all-ones around every WMMA).
    const int wid  = __builtin_amdgcn_readfirstlane(
                        blockIdx.x * 8 + (threadIdx.x >> 5));  // 0..2047
    const int b    = wid >> 9;                                 // /512
    const int r    = wid & 511;
    const int pair = r >> 1;                                   // 0..255
    const int s    = r & 1;                                    // column parity
    const int l16  = lane & 15;
    const int hi   = lane >> 4;                                // lane half

    const float4* pb = packed + (size_t)b * NPTS;
    const float*  wb = wbuf   + (size_t)b * NPTS;

    float acc1 = 0.0f;   // diagonal tiles (weight 1)
    float acc2 = 0.0f;   // strictly-upper tiles (weight 2)

#pragma unroll
    for (int pass = 0; pass < 2; ++pass) {
        const int rt = pass ? (NTILE - 1 - pair) : pair;       // row tile
        const int n0 = rt * 16;

        // A matrix: 16x4 tile (rows n0..n0+15, K=3 padded).
        // ISA A-layout: lanes 0-15 hold K=0,1; lanes 16-31 hold K=2,3.
        const float4 ap = pb[n0 + l16];
        v2f av;
        av.x = hi ? ap.z : ap.x;
        av.y = hi ? 0.0f : ap.y;

        // Per-row norms/weights. D-layout: VGPR v -> M = v + 8*hi.
        float sqn[8], wn[8];
#pragma unroll
        for (int v = 0; v < 8; ++v) {
            const int row = n0 + v + hi * 8;
            sqn[v] = pb[row].w;
            wn[v]  = wb[row];
        }

        // Parity s=0 owns the diagonal tile + rt+2, rt+4, ...
        // Parity s=1 owns rt+1, rt+3, ...
        if (s == 0)
            acc1 += KUN_tile(pb, wb, av, sqn, wn, rt, l16, hi);

#pragma unroll 2
        for (int mt = rt + 2 - s; mt < NTILE; mt += 2) {
            if (mt + 2 < NTILE)
                __builtin_prefetch(&pb[(mt + 2) * 16 + l16], 0, 1); // global_prefetch_b8
            acc2 += KUN_tile(pb, wb, av, sqn, wn, mt, l16, hi);
        }
    }

    float acc = fmaf(2.0f, acc2, acc1);

    // Wave32 reduction, then one atomic per wave (2048 atomics total).
#pragma unroll
    for (int off = 16; off > 0; off >>= 1)
        acc += __shfl_down(acc, off, 32);
    if (lane == 0) {
        const float scale = LAMBDA / ((float)NB * (float)NPTS * (float)NPTS);
        atomicAdd(out, acc * scale);
    }
}

extern "C" void kernel_launch(void* const* d_in, const int* in_sizes, int n_in,
                              void* d_out, int out_size, void* d_ws, size_t ws_size,
                              hipStream_t stream) {
    (void)in_sizes; (void)n_in; (void)out_size; (void)ws_size;
    const float* inputs = (const float*)d_in[0];   // (4, 2, 8192) f32
    const float* coords = (const float*)d_in[1];   // (4, 8192, 3) f32
    float* out = (float*)d_out;                    // 1 x f32

    // Workspace layout: [0, 512KB) packed float4; [512KB, 640KB) weights.
    float4* packed = (float4*)d_ws;
    float*  wbuf   = (float*)((char*)d_ws + (size_t)NPTOT * sizeof(float4));

    KUN_dpp_pack<<<(NPTOT + 255) / 256, 256, 0, stream>>>(inputs, coords,
                                                          packed, wbuf, out);
    // 2048 waves = 4 batches * 256 row-tile pairs * 2 column parities.
    KUN_dpp_kernel<<<256, 256, 0, stream>>>(packed, wbuf, out);
}